// LQRSolver_52862457479963
// MI455X (gfx1250) — compile-verified
//
#include <hip/hip_runtime.h>

// LQR backward pass (Riccati recursion) for MI455X / gfx1250.
// N_STATE=512, N_CTRL=128, DIM=640, HORIZON=256. All math fp32 via
// V_WMMA_F32_16X16X4_F32 (CDNA5 fp32 matrix pipe).

#define NS 512
#define NC 128
#define DIMN 640
#define T_HOR 256

typedef __attribute__((ext_vector_type(2))) float v2f;
typedef __attribute__((ext_vector_type(8))) float v8f;

// ---------------------------------------------------------------------------
// Generic tiled fp32 WMMA GEMM:  D = alpha * op(A) @ B  (+ Csrc if non-null)
// op(A) = A (TRANSA=false, A is M x K row-major) or A^T (TRANSA=true, A is
// K x M row-major). Block tile 128(M) x 128(N), BK=16. 256 threads = 8 waves,
// each wave owns a 32x64 output tile = 2x4 WMMA 16x16 accumulators.
// LDS is fragment-native: As[m][k], Bs[n][k] with row length 18 (even pad)
// so every WMMA fragment is a single aligned ds_load_b64 (v2f).
// Requires M%128==0, N%128==0, K%16==0 (true for all shapes here).
// ---------------------------------------------------------------------------
#define LROW 18  // LDS row length in floats (16 + 2 pad; even => b64 aligned)

template <bool TRANSA>
__global__ __launch_bounds__(256) void wmma_gemm_f32(
    const float* __restrict__ A, int lda,
    const float* __restrict__ B, int ldb,
    const float* __restrict__ Csrc, int ldc,
    float* __restrict__ D, int ldd,
    int Kdim, float alpha) {
  __shared__ float As[128 * LROW];  // [m][k]
  __shared__ float Bs[128 * LROW];  // [n][k]

  const int t  = threadIdx.x;
  const int m0 = blockIdx.y * 128;
  const int n0 = blockIdx.x * 128;
  const int lane = t & 31;
  const int wid  = t >> 5;
  const int wm   = (wid >> 1) * 32;  // wave row offset within block tile
  const int wn   = (wid & 1) * 64;   // wave col offset within block tile
  const int l16  = lane & 15;
  const int hi   = lane >> 4;        // half-wave: K pair 0/1 vs 2/3, M r vs r+8

  const v8f vzero = {0.f, 0.f, 0.f, 0.f, 0.f, 0.f, 0.f, 0.f};
  v8f acc[2][4];
#pragma unroll
  for (int i = 0; i < 2; ++i)
#pragma unroll
    for (int j = 0; j < 4; ++j) acc[i][j] = vzero;

  for (int k0 = 0; k0 < Kdim; k0 += 16) {
    if (k0 + 16 < Kdim) {  // pull next K-tile toward the caches
      __builtin_prefetch(&B[(size_t)(k0 + 16) * ldb + n0], 0, 1);
      if (TRANSA)
        __builtin_prefetch(&A[(size_t)(k0 + 16) * lda + m0], 0, 1);
      else
        __builtin_prefetch(&A[(size_t)m0 * lda + (k0 + 16)], 0, 1);
    }
    // Stage op(A) 128x16 tile as As[m][k]
    if (TRANSA) {
#pragma unroll
      for (int i = 0; i < 8; ++i) {
        int idx = t + i * 256;
        int m = idx & 127, kk = idx >> 7;         // coalesced in m
        As[m * LROW + kk] = A[(size_t)(k0 + kk) * lda + (m0 + m)];
      }
    } else {
#pragma unroll
      for (int i = 0; i < 8; ++i) {
        int idx = t + i * 256;
        int kk = idx & 15, m = idx >> 4;          // coalesced in k (16-wide)
        As[m * LROW + kk] = A[(size_t)(m0 + m) * lda + (k0 + kk)];
      }
    }
    // Stage B 16x128 tile as Bs[n][k]
#pragma unroll
    for (int i = 0; i < 8; ++i) {
      int idx = t + i * 256;
      int n = idx & 127, kk = idx >> 7;           // coalesced in n
      Bs[n * LROW + kk] = B[(size_t)(k0 + kk) * ldb + (n0 + n)];
    }
    __syncthreads();

#pragma unroll
    for (int kk = 0; kk < 4; ++kk) {
      const int kc = kk * 4 + 2 * hi;  // this half-wave's K pair
      v2f a[2], b[4];
#pragma unroll
      for (int im = 0; im < 2; ++im)
        a[im] = *(const v2f*)&As[(wm + im * 16 + l16) * LROW + kc];
#pragma unroll
      for (int in = 0; in < 4; ++in)
        b[in] = *(const v2f*)&Bs[(wn + in * 16 + l16) * LROW + kc];
#pragma unroll
      for (int im = 0; im < 2; ++im)
#pragma unroll
        for (int in = 0; in < 4; ++in)
          acc[im][in] = __builtin_amdgcn_wmma_f32_16x16x4_f32(
              false, a[im], false, b[in], (short)0, acc[im][in], false, false);
    }
    __syncthreads();
  }

  // Epilogue: C/D layout — VGPR r: lanes 0-15 -> row r, lanes 16-31 -> row r+8
#pragma unroll
  for (int im = 0; im < 2; ++im)
#pragma unroll
    for (int in = 0; in < 4; ++in) {
      const int col = n0 + wn + in * 16 + l16;
#pragma unroll
      for (int r = 0; r < 8; ++r) {
        const int row = m0 + wm + im * 16 + hi * 8 + r;
        float val = alpha * acc[im][in][r];
        if (Csrc) val += Csrc[(size_t)row * ldc + col];
        D[(size_t)row * ldd + col] = val;
      }
    }
}

// ---------------------------------------------------------------------------
// q = c + FV @ f + F^T @ v          (DIMN vector)
// ---------------------------------------------------------------------------
__global__ void qvec_kernel(const float* __restrict__ F,
                            const float* __restrict__ FV,
                            const float* __restrict__ cvec,
                            const float* __restrict__ fvec,
                            const float* __restrict__ vin,
                            float* __restrict__ q) {
  int i = blockIdx.x * blockDim.x + threadIdx.x;
  if (i >= DIMN) return;
  float s = cvec[i];
  const float* fvrow = FV + (size_t)i * NS;
  for (int k = 0; k < NS; ++k) s += fvrow[k] * fvec[k];
  for (int k = 0; k < NS; ++k) s += F[(size_t)k * DIMN + i] * vin[k];
  q[i] = s;
}

// ---------------------------------------------------------------------------
// In-place Gauss-Jordan inversion of 128x128 SPD Quu, one workgroup, LDS.
// Dynamic LDS: 128*129 matrix (pad to kill bank conflicts) + 128 aux.
// ---------------------------------------------------------------------------
__global__ __launch_bounds__(256) void gj_invert128(
    const float* __restrict__ Quu, int ld, float* __restrict__ inv) {
  extern __shared__ float S[];
  float* aux = S + 128 * 129;
  const int t = threadIdx.x;

  for (int idx = t; idx < 128 * 128; idx += 256) {
    int r = idx >> 7, c = idx & 127;
    S[r * 129 + c] = Quu[(size_t)r * ld + c];
  }
  __syncthreads();

  for (int p = 0; p < 128; ++p) {
    const float invd = 1.0f / S[p * 129 + p];
    if (t < 128) aux[t] = S[t * 129 + p];  // snapshot column p
    __syncthreads();
    if (t < 128) {  // scale pivot row; A[p][p] <- 1/d
      float val = S[p * 129 + t] * invd;
      if (t == p) val = invd;
      S[p * 129 + t] = val;
    }
    __syncthreads();
    {  // eliminate: each thread owns half a row
      const int i = t >> 1;
      const int cb = (t & 1) << 6;
      if (i != p) {
        const float fm = aux[i];
        for (int cc = 0; cc < 64; ++cc) {
          const int col = cb + cc;
          float val = S[i * 129 + col] - fm * S[p * 129 + col];
          if (col == p) val = -fm * invd;
          S[i * 129 + col] = val;
        }
      }
    }
    __syncthreads();
  }

  for (int idx = t; idx < 128 * 128; idx += 256) {
    int r = idx >> 7, c = idx & 127;
    inv[idx] = S[r * 129 + c];
  }
}

// k = -Quu_inv @ qu
__global__ void kvec_kernel(const float* __restrict__ Qinv,
                            const float* __restrict__ q,
                            float* __restrict__ kout) {
  int i = threadIdx.x;
  const float* row = Qinv + (size_t)i * NC;
  float s = 0.f;
  for (int j = 0; j < NC; ++j) s += row[j] * q[NS + j];
  kout[i] = -s;
}

// vn = qx + P @ k + K^T @ qu     (P = Qxu + K^T Quu)
__global__ void vnvec_kernel(const float* __restrict__ P,
                             const float* __restrict__ Kmat,
                             const float* __restrict__ q,
                             const float* __restrict__ kvec,
                             float* __restrict__ vout) {
  int i = blockIdx.x * blockDim.x + threadIdx.x;  // < NS
  float s = q[i];
  const float* Prow = P + (size_t)i * NC;
  for (int j = 0; j < NC; ++j) s += Prow[j] * kvec[j];
  for (int j = 0; j < NC; ++j) s += Kmat[(size_t)j * NS + i] * q[NS + j];
  vout[i] = s;
}

// const_n = const_prev + 0.5 k.(Quu k) + k.qu + 0.5 f.(V f) + f.v
__global__ __launch_bounds__(256) void const_kernel(
    const float* __restrict__ Qfull, const float* __restrict__ q,
    const float* __restrict__ kvec, const float* __restrict__ fvec,
    const float* __restrict__ Vin, int ldV, const float* __restrict__ vin,
    const float* __restrict__ cin, float* __restrict__ outc) {
  __shared__ float red[256];
  const int t = threadIdx.x;
  float s = 0.f;
  if (t < NC) {
    const float* Qrow = Qfull + (size_t)(NS + t) * DIMN + NS;  // Quu row t
    float acc = 0.f;
    for (int j = 0; j < NC; ++j) acc += Qrow[j] * kvec[j];
    s += kvec[t] * (0.5f * acc + q[NS + t]);
  }
  for (int r = t; r < NS; r += 256) {
    const float* Vrow = Vin + (size_t)r * ldV;
    float acc = 0.f;
    for (int j = 0; j < NS; ++j) acc += Vrow[j] * fvec[j];
    s += fvec[r] * (0.5f * acc + vin[r]);
  }
  red[t] = s;
  __syncthreads();
  for (int off = 128; off > 0; off >>= 1) {
    if (t < off) red[t] += red[t + off];
    __syncthreads();
  }
  if (t == 0) *outc = *cin + red[0];
}

__global__ void zero_kernel(float* p) {
  if (threadIdx.x == 0) *p = 0.f;
}

// ---------------------------------------------------------------------------
extern "C" void kernel_launch(void* const* d_in, const int* in_sizes, int n_in,
                              void* d_out, int out_size, void* d_ws,
                              size_t ws_size, hipStream_t stream) {
  const float* F  = (const float*)d_in[0];  // 512 x 640
  const float* fv = (const float*)d_in[1];  // 512
  const float* C  = (const float*)d_in[2];  // 640 x 640
  const float* cv = (const float*)d_in[3];  // 640
  (void)in_sizes; (void)n_in; (void)out_size; (void)ws_size;

  float* out = (float*)d_out;
  float* ws  = (float*)d_ws;

  // Workspace layout (floats)
  float* FV   = ws;                           // 640*512
  float* Q    = FV + (size_t)DIMN * NS;       // 640*640
  float* q    = Q + (size_t)DIMN * DIMN;      // 640
  float* Qinv = q + DIMN;                     // 128*128
  float* P    = Qinv + (size_t)NC * NC;       // 512*128
  float* zc   = P + (size_t)NS * NC;          // 1 (const seed = 0)

  // Output layout (concatenated, reversed time): Ks, ks, Vs, vs, consts
  const size_t KS_OFF = 0;
  const size_t ks_OFF = (size_t)T_HOR * NC * NS;
  const size_t VS_OFF = ks_OFF + (size_t)T_HOR * NC;
  const size_t vs_OFF = VS_OFF + (size_t)T_HOR * NS * NS;
  const size_t cs_OFF = vs_OFF + (size_t)T_HOR * NS;

  zero_kernel<<<1, 32, 0, stream>>>(zc);

  for (int t = 0; t < T_HOR; ++t) {
    const float* Vin; int ldV; const float* vin; const float* cin;
    if (t == 0) {
      Vin = C; ldV = DIMN; vin = cv; cin = zc;       // V0 = C[:512,:512]
    } else {
      const int tp = T_HOR - t;                      // previous step's out slot
      Vin = out + VS_OFF + (size_t)tp * NS * NS; ldV = NS;
      vin = out + vs_OFF + (size_t)tp * NS;
      cin = out + cs_OFF + tp;
    }
    const int to = T_HOR - 1 - t;
    float* Kout = out + KS_OFF + (size_t)to * NC * NS;
    float* kout = out + ks_OFF + (size_t)to * NC;
    float* Vout = out + VS_OFF + (size_t)to * NS * NS;
    float* vout = out + vs_OFF + (size_t)to * NS;
    float* cout = out + cs_OFF + to;

    // 1) FV = F^T @ V                    (640 x 512, K=512)
    wmma_gemm_f32<true><<<dim3(NS / 128, DIMN / 128), 256, 0, stream>>>(
        F, DIMN, Vin, ldV, nullptr, 0, FV, NS, NS, 1.0f);
    // 2) q = c + FV@f + F^T@v
    qvec_kernel<<<DIMN / 128, 128, 0, stream>>>(F, FV, cv, fv, vin, q);
    // 3) Q = C + FV @ F                  (640 x 640, K=512)
    wmma_gemm_f32<false><<<dim3(DIMN / 128, DIMN / 128), 256, 0, stream>>>(
        FV, NS, F, DIMN, C, DIMN, Q, DIMN, NS, 1.0f);
    // 4) Qinv = inv(Quu)                 (single WGP, LDS Gauss-Jordan)
    gj_invert128<<<1, 256, (128 * 129 + 128) * sizeof(float), stream>>>(
        Q + (size_t)NS * DIMN + NS, DIMN, Qinv);
    // 5) K = -Qinv @ Qux                 (128 x 512, K=128)
    wmma_gemm_f32<false><<<dim3(NS / 128, NC / 128), 256, 0, stream>>>(
        Qinv, NC, Q + (size_t)NS * DIMN, DIMN, nullptr, 0, Kout, NS, NC, -1.0f);
    // 6) k = -Qinv @ qu
    kvec_kernel<<<1, NC, 0, stream>>>(Qinv, q, kout);
    // 7) P = Qxu + K^T @ Quu             (512 x 128, K=128)
    wmma_gemm_f32<true><<<dim3(NC / 128, NS / 128), 256, 0, stream>>>(
        Kout, NS, Q + (size_t)NS * DIMN + NS, DIMN, Q + NS, DIMN, P, NC, NC, 1.0f);
    // 8) Vn = Qxx + P @ K                (512 x 512, K=128)
    wmma_gemm_f32<false><<<dim3(NS / 128, NS / 128), 256, 0, stream>>>(
        P, NC, Kout, NS, Q, DIMN, Vout, NS, NC, 1.0f);
    // 9) Vn += K^T @ Qux                 (512 x 512, K=128)
    wmma_gemm_f32<true><<<dim3(NS / 128, NS / 128), 256, 0, stream>>>(
        Kout, NS, Q + (size_t)NS * DIMN, DIMN, Vout, NS, Vout, NS, NC, 1.0f);
    // 10) vn = qx + P@k + K^T@qu
    vnvec_kernel<<<NS / 128, 128, 0, stream>>>(P, Kout, q, kout, vout);
    // 11) const update (uses OLD V, v, const)
    const_kernel<<<1, 256, 0, stream>>>(Q, q, kout, fv, Vin, ldV, vin, cin, cout);
  }
}